// LinearSystemStepsize_52956946759998
// MI455X (gfx1250) — compile-verified
//
#include <hip/hip_runtime.h>
#include <hip/hip_bf16.h>

// Problem constants (from reference).
#define L_STEPS 2048
#define BATCH   256
#define NN      64
#define MM      64
#define UCHUNK  128   // u-preload chunk (steps); must divide L_STEPS

typedef float v2f __attribute__((ext_vector_type(2)));
typedef float v8f __attribute__((ext_vector_type(8)));

// ---------------------------------------------------------------------------
// Kernel 1: per-step Gauss-Jordan precompute.
//   Solve (I - h A) [T | b] = [(I + h A) | dt*B] for each time step t.
//   Output: wsT[t][n][k] = T[n][k] (64x64), wsb[t][n] = b[n].
//   A = 0.5(G - G^T) - I => (I - hA) is well conditioned with dominant
//   diagonal (1+h): pivot-free Gauss-Jordan is fine.
// ---------------------------------------------------------------------------
__global__ __launch_bounds__(256) void lss_precompute_kernel(
    const float* __restrict__ dt, const float* __restrict__ A,
    const float* __restrict__ Bv, float* __restrict__ wsT,
    float* __restrict__ wsb) {
  __shared__ float aug[64][132];  // 64 rows x 129 cols, padded
  const int t = blockIdx.x;
  const float dtv = dt[t];
  const float h = 0.5f * dtv;
  const int tid = threadIdx.x;

  // Build augmented matrix [ (I - hA) | (I + hA) | dt*B ].
  for (int idx = tid; idx < 64 * 129; idx += 256) {
    const int r = idx / 129, c = idx % 129;
    float v;
    if (c < 64) {
      v = (r == c ? 1.0f : 0.0f) - h * A[r * 64 + c];
    } else if (c < 128) {
      const int c2 = c - 64;
      v = (r == c2 ? 1.0f : 0.0f) + h * A[r * 64 + c2];
    } else {
      v = dtv * Bv[r];
    }
    aug[r][c] = v;
  }
  __syncthreads();

  // 4 threads per row; each owns 33 columns.
  const int r = tid >> 2;
  const int cg = tid & 3;
  for (int p = 0; p < 64; ++p) {
    const float inv = 1.0f / aug[p][p];
    const float fac = aug[r][p] * inv;  // read before any writes this round
    __syncthreads();
    if (r != p) {
      for (int j = 0; j < 33; ++j) {
        const int c = cg * 33 + j;
        if (c < 129) aug[r][c] -= fac * aug[p][c];
      }
    }
    __syncthreads();
    if (r == p) {
      for (int j = 0; j < 33; ++j) {
        const int c = cg * 33 + j;
        if (c < 129) aug[p][c] *= inv;
      }
    }
    __syncthreads();
  }

  // Write out T (cols 64..127) and b (col 128).
  for (int idx = tid; idx < 64 * 64; idx += 256) {
    const int n = idx >> 6, k = idx & 63;
    wsT[(size_t)t * 4096 + idx] = aug[n][64 + k];
  }
  if (tid < 64) wsb[t * 64 + tid] = aug[tid][128];
}

// ---------------------------------------------------------------------------
// Kernel 2: the sequential scan, x-update ONLY (y is computed later, off the
// critical path). Grid = 16 blocks (16 batch rows each), 128 threads = 4
// waves; wave w owns column block w.
//   x_t = x_{t-1} @ T_t^T + u_t (x) b_t
// K=64 reduction split into two independent 8-WMMA chains for XDL pipelining.
// x_t is streamed into the y region of d_out (N == M == 64), to be
// transformed in place by kernel 3.
// ---------------------------------------------------------------------------
__global__ __launch_bounds__(128) void lss_scan_kernel(
    const float* __restrict__ u, const float* __restrict__ wsT,
    const float* __restrict__ wsb, float* __restrict__ xseq,
    float* __restrict__ x_final) {
  __shared__ float xs[16 * 66];      // x chunk, row stride 66 (bank pad)
  __shared__ float ub[UCHUNK * 16];  // u preload: 128 steps x 16 batch rows

  const int tid = threadIdx.x;
  const int lane = tid & 31;
  const int w = tid >> 5;      // wave id 0..3 -> column block
  const int half = lane >> 4;  // 0: lanes 0-15, 1: lanes 16-31
  const int col = lane & 15;   // column within block; also A-operand row
  const int b0 = blockIdx.x * 16;
  const int mrow = w * 16 + col;

  // x_0 = 0
  for (int i = tid; i < 16 * 66; i += 128) xs[i] = 0.0f;
  __syncthreads();

  for (int t = 0; t < L_STEPS; ++t) {
    // Refill u chunk every UCHUNK steps (amortized: 2 barriers / 128 steps).
    if ((t & (UCHUNK - 1)) == 0) {
      __syncthreads();  // prior reads of ub complete
      for (int i = tid; i < UCHUNK * 16; i += 128)
        ub[i] = u[(t + (i >> 4)) * BATCH + b0 + (i & 15)];
      __syncthreads();
    }

    const float* Tt = wsT + (size_t)t * 4096;
    const float bval = wsb[t * 64 + mrow];
    const int tl = (t & (UCHUNK - 1)) * 16;

    float ur[8];
#pragma unroll
    for (int r = 0; r < 8; ++r) ur[r] = ub[tl + half * 8 + r];

    // Two independent accumulator chains; acc0 seeded with u (x) b.
    v8f acc0, acc1;
#pragma unroll
    for (int r = 0; r < 8; ++r) {
      acc0[r] = ur[r] * bval;
      acc1[r] = 0.0f;
    }

    // Prefetch next step's T rows for this wave's column block (L2 hint).
    if (t + 1 < L_STEPS)
      __builtin_prefetch(wsT + (size_t)(t + 1) * 4096 + mrow * 64, 0, 1);

    // x_new[:, w-block] = x @ T^T : B-op[kk][n'] = T[n0+n'][k+kk].
#pragma unroll
    for (int kb = 0; kb < 8; ++kb) {
      const int k0 = kb * 4 + half * 2;
      const int k1 = 32 + k0;
      const v2f a0 = *(const v2f*)&xs[col * 66 + k0];
      const v2f t0 = *(const v2f*)&Tt[mrow * 64 + k0];
      const v2f a1 = *(const v2f*)&xs[col * 66 + k1];
      const v2f t1 = *(const v2f*)&Tt[mrow * 64 + k1];
      acc0 = __builtin_amdgcn_wmma_f32_16x16x4_f32(
          false, a0, false, t0, (short)0, acc0, false, false);
      acc1 = __builtin_amdgcn_wmma_f32_16x16x4_f32(
          false, a1, false, t1, (short)0, acc1, false, false);
    }
    __syncthreads();  // all reads of x_{t-1} done

    // Merge chains; write x_t to LDS and stream it to global (y slot).
    float* xt = xseq + ((size_t)t * BATCH + b0) * MM;
#pragma unroll
    for (int r = 0; r < 8; ++r) {
      const float v = acc0[r] + acc1[r];
      const int row = r + half * 8;
      xs[row * 66 + mrow] = v;
      xt[row * MM + mrow] = v;  // fire-and-forget store
    }
    __syncthreads();  // xs now holds x_t
  }

  // x_final[b0+i][n] from LDS.
  for (int idx = tid; idx < 16 * 64; idx += 128) {
    const int i = idx >> 6, n = idx & 63;
    x_final[(size_t)(b0 + i) * NN + n] = xs[i * 66 + n];
  }
}

// ---------------------------------------------------------------------------
// Kernel 3: fully parallel in-place map  y[t] = x[t] @ C^T + u[t] (x) D.
// Grid = (2048, 16) workgroups of 128 threads; each WG owns a 16x64 chunk of
// d_out: stage x chunk in LDS, barrier, then overwrite with y. Full-chip
// occupancy WMMA GEMM (~4.3 GFLOP over ~270 MB of traffic).
// ---------------------------------------------------------------------------
__global__ __launch_bounds__(128) void lss_ymap_kernel(
    const float* __restrict__ u, const float* __restrict__ C,
    const float* __restrict__ D, float* y) {
  __shared__ float xs[16 * 66];

  const int tid = threadIdx.x;
  const int lane = tid & 31;
  const int w = tid >> 5;
  const int half = lane >> 4;
  const int col = lane & 15;
  const int t = blockIdx.x;
  const int b0 = blockIdx.y * 16;
  const int mrow = w * 16 + col;

  float* chunk = y + ((size_t)t * BATCH + b0) * MM;

  // Stage x chunk (coalesced) into LDS with padded stride.
  for (int idx = tid; idx < 16 * 64; idx += 128) {
    xs[(idx >> 6) * 66 + (idx & 63)] = chunk[idx];
  }

  // B-operands for C^T: B[kk][m'] = C[m0+m'][k+kk] (contiguous pair / lane).
  v2f cB[16];
#pragma unroll
  for (int kb = 0; kb < 16; ++kb) {
    cB[kb] = *(const v2f*)&C[mrow * 64 + kb * 4 + half * 2];
  }
  const float dval = D[mrow];

  float ur[8];
#pragma unroll
  for (int r = 0; r < 8; ++r) ur[r] = u[t * BATCH + b0 + half * 8 + r];

  v8f acc0, acc1;
#pragma unroll
  for (int r = 0; r < 8; ++r) {
    acc0[r] = ur[r] * dval;  // skip connection u (x) D
    acc1[r] = 0.0f;
  }
  __syncthreads();  // x chunk fully staged before any overwrite

#pragma unroll
  for (int kb = 0; kb < 8; ++kb) {
    const int k0 = kb * 4 + half * 2;
    const int k1 = 32 + k0;
    const v2f a0 = *(const v2f*)&xs[col * 66 + k0];
    const v2f a1 = *(const v2f*)&xs[col * 66 + k1];
    acc0 = __builtin_amdgcn_wmma_f32_16x16x4_f32(
        false, a0, false, cB[kb], (short)0, acc0, false, false);
    acc1 = __builtin_amdgcn_wmma_f32_16x16x4_f32(
        false, a1, false, cB[kb + 8], (short)0, acc1, false, false);
  }

  // Overwrite chunk with y.
#pragma unroll
  for (int r = 0; r < 8; ++r) {
    chunk[(r + half * 8) * MM + mrow] = acc0[r] + acc1[r];
  }
}

// ---------------------------------------------------------------------------
// Host-side launcher.
// Inputs (setup_inputs order): dt(L), u(L*BATCH), A(N*N), B(N), C(M*N), D(M).
// Output: y (L*BATCH*M floats) followed by x_final (BATCH*N floats).
// Workspace: wsT = 2048*4096 floats, wsb = 2048*64 floats (~34 MB).
// ---------------------------------------------------------------------------
extern "C" void kernel_launch(void* const* d_in, const int* in_sizes, int n_in,
                              void* d_out, int out_size, void* d_ws,
                              size_t ws_size, hipStream_t stream) {
  const float* dt = (const float*)d_in[0];
  const float* u  = (const float*)d_in[1];
  const float* A  = (const float*)d_in[2];
  const float* Bv = (const float*)d_in[3];
  const float* C  = (const float*)d_in[4];
  const float* D  = (const float*)d_in[5];

  float* wsT = (float*)d_ws;
  float* wsb = wsT + (size_t)L_STEPS * NN * NN;

  float* y_out   = (float*)d_out;
  float* x_final = y_out + (size_t)L_STEPS * BATCH * MM;

  // 1) Per-step T_t, b_t (fully parallel, 2048 WGs).
  lss_precompute_kernel<<<L_STEPS, 256, 0, stream>>>(dt, A, Bv, wsT, wsb);
  // 2) Sequential scan: x_t streamed into the y region of d_out.
  lss_scan_kernel<<<BATCH / 16, 128, 0, stream>>>(u, wsT, wsb, y_out,
                                                  x_final);
  // 3) In-place x -> y transform, full-chip WMMA GEMM.
  lss_ymap_kernel<<<dim3(L_STEPS, BATCH / 16), 128, 0, stream>>>(u, C, D,
                                                                 y_out);
}